// SSD_67534065762863
// MI455X (gfx1250) — compile-verified
//
#include <hip/hip_runtime.h>
#include <stdint.h>

// ---------------- problem constants (match reference) ----------------
#define B_      128
#define D_      8732
#define C_      21
#define TOPK    200
#define CONF_T  0.01f
#define NMS_T   0.45f
#define NTHR    256

typedef __attribute__((ext_vector_type(4))) unsigned int v4u;
typedef __attribute__((ext_vector_type(8))) int          v8i;
typedef __attribute__((ext_vector_type(4))) int          v4i;

#if defined(__has_builtin)
#if __has_builtin(__builtin_amdgcn_tensor_load_to_lds)
#define HAVE_TDM 1
#endif
#endif
#ifndef HAVE_TDM
#define HAVE_TDM 0
#endif

// ---------------- Kernel A: per-anchor softmax stats ----------------
// For each (b,d): mx = max_c conf, invden = 1 / sum_c exp(conf - mx)
__global__ void ssd_stats_kernel(const float* __restrict__ conf,
                                 float* __restrict__ mx,
                                 float* __restrict__ invden, int n) {
  int i = blockIdx.x * blockDim.x + threadIdx.x;
  if (i >= n) return;
  const float* p = conf + (size_t)i * C_;
  __builtin_prefetch(p + (size_t)C_ * NTHR, 0, 0);  // global_prefetch_b8
  float m = p[0];
#pragma unroll
  for (int c = 1; c < C_; ++c) m = fmaxf(m, p[c]);
  float s = 0.0f;
#pragma unroll
  for (int c = 0; c < C_; ++c) s += __expf(p[c] - m);
  mx[i] = m;
  invden[i] = 1.0f / s;
}

// ---------------- Kernel B: per-(image,class) top-K + NMS ----------------
__global__ void __launch_bounds__(NTHR)
ssd_nms_kernel(const float* __restrict__ conf,
               const float* __restrict__ loc,
               const float* __restrict__ dbox,
               const float* __restrict__ mx,
               const float* __restrict__ invden,
               float* __restrict__ out) {
  __shared__ uint32_t sKey[D_];                 // conf column, then score keys
  __shared__ uint32_t sHist[NTHR];              // radix histogram / scan buf
  __shared__ unsigned long long sCand[NTHR];    // (key<<32)|anchor, sorted desc
  __shared__ float sBox[NTHR][4];               // decoded candidate boxes
  __shared__ int sKeep[NTHR];
  __shared__ int sCtl[8];  // 0:prefix 1:Krem 2:done 4:threshold 5:gtCtr 6:eqCtr

  const int tid = threadIdx.x;
  const int b   = blockIdx.x / (C_ - 1);
  const int cls = blockIdx.x % (C_ - 1) + 1;    // skip background

  // ---- stage 1: DMA conf[b, :, cls] (stride 21 floats) into LDS via TDM ----
#if HAVE_TDM
  if ((tid >> 5) == 0) {  // wave 0 issues the tensor DMA
    uint64_t ga  = (uint64_t)(uintptr_t)(conf + ((size_t)b * D_ * C_ + (size_t)cls));
    uint32_t lds = (uint32_t)(uintptr_t)&sKey[0];

    v4u g0;
    g0.x = 1u;                                            // count=1, user mode
    g0.y = lds;                                           // lds_addr
    g0.z = (uint32_t)(ga & 0xFFFFFFFFu);                  // global_addr[31:0]
    g0.w = (uint32_t)((ga >> 32) & 0x01FFFFFFu) | (2u << 30);  // [56:32], type=2

    const uint32_t dim0 = (uint32_t)C_;    // tensor x extent (cols)
    const uint32_t dim1 = (uint32_t)D_;    // tensor y extent (rows)
    const uint32_t str0 = (uint32_t)C_;    // row stride in elements
    v8i g1;
    g1[0] = (int)(2u << 16);                                   // data_size=4B
    g1[1] = (int)((dim0 & 0xFFFFu) << 16);                     // dim0 lo16
    g1[2] = (int)((dim0 >> 16) | ((dim1 & 0xFFFFu) << 16));    // dim0 hi|dim1 lo
    g1[3] = (int)((dim1 >> 16) | (1u << 16));                  // dim1 hi, tile_dim0=1
    g1[4] = (int)(D_ & 0xFFFF);                                // tile_dim1=8732
    g1[5] = (int)str0;                                         // dim0_stride lo32
    g1[6] = 0;
    g1[7] = 0;
    v4i g2 = {0, 0, 0, 0};
    v4i g3 = {0, 0, 0, 0};
    v8i g4 = {0, 0, 0, 0, 0, 0, 0, 0};
    __builtin_amdgcn_tensor_load_to_lds(g0, g1, g2, g3, g4, 0);
    __builtin_amdgcn_s_wait_tensorcnt(0);
  }
#else
  for (int d = tid; d < D_; d += NTHR)
    sKey[d] = __float_as_uint(conf[(size_t)b * D_ * C_ + (size_t)d * C_ + cls]);
#endif
  __syncthreads();

  // ---- stage 2: logits -> softmax score -> monotonic uint key (0 = masked) ----
  {
    const float* mxp = mx + (size_t)b * D_;
    const float* idp = invden + (size_t)b * D_;
    for (int d = tid; d < D_; d += NTHR) {
      float cr = __uint_as_float(sKey[d]);
      float s  = __expf(cr - mxp[d]) * idp[d];       // softmax score in (0,1]
      sKey[d]  = (s > CONF_T) ? __float_as_uint(s) : 0u;  // bits monotonic
    }
  }
  if (tid == 0) { sCtl[0] = 0; sCtl[1] = TOPK; sCtl[2] = 0; sCtl[4] = 0; }
  __syncthreads();

  // ---- stage 3: MSB-first radix select of the TOPK-th largest key ----
  for (int p = 3; p >= 0; --p) {
    sHist[tid] = 0u;
    __syncthreads();
    if (!sCtl[2]) {
      uint32_t prefix = (uint32_t)sCtl[0];
      for (int d = tid; d < D_; d += NTHR) {
        uint32_t key = sKey[d];
        if (key != 0u &&
            (uint32_t)(((uint64_t)key) >> ((p + 1) * 8)) == prefix)
          atomicAdd(&sHist[(key >> (p * 8)) & 0xFFu], 1u);
      }
    }
    __syncthreads();
    if (tid == 0 && !sCtl[2]) {
      int K = sCtl[1];
      if (p == 3) {  // total valid count: maybe fewer than TOPK candidates
        uint32_t tot = 0;
        for (int i = 0; i < NTHR; ++i) tot += sHist[i];
        if ((int)tot <= K) { sCtl[4] = 0; sCtl[2] = 1; }
      }
      if (!sCtl[2]) {
        uint32_t cum = 0; int bin = 0;
        for (int i = 255; i >= 0; --i) {
          if ((int)(cum + sHist[i]) >= K) { bin = i; break; }
          cum += sHist[i];
        }
        sCtl[0] = (sCtl[0] << 8) | bin;
        sCtl[1] = K - (int)cum;
        if (p == 0) sCtl[4] = sCtl[0];   // exact 32-bit threshold key
      }
    }
    __syncthreads();
  }

  // ---- stage 4: compact candidates (key > T, then ties == T) ----
  if (tid == 0) { sCtl[5] = 0; sCtl[6] = 0; }
  sCand[tid] = 0ull;
  __syncthreads();
  const uint32_t T = (uint32_t)sCtl[4];
  for (int d = tid; d < D_; d += NTHR) {
    uint32_t key = sKey[d];
    if (key > T) {
      int pos = atomicAdd(&sCtl[5], 1);
      if (pos < TOPK) sCand[pos] = ((unsigned long long)key << 32) | (unsigned)d;
    }
  }
  __syncthreads();
  int nGT = sCtl[5]; if (nGT > TOPK) nGT = TOPK;
  if (T != 0u) {
    int room = TOPK - nGT;
    for (int d = tid; d < D_; d += NTHR) {
      if (sKey[d] == T) {
        int pos = atomicAdd(&sCtl[6], 1);
        if (pos < room)
          sCand[nGT + pos] = ((unsigned long long)T << 32) | (unsigned)d;
      }
    }
  }
  __syncthreads();

  // ---- stage 5: bitonic sort 256 keys descending (pads are 0) ----
  for (int k = 2; k <= NTHR; k <<= 1) {
    for (int j = k >> 1; j > 0; j >>= 1) {
      int i = tid, ixj = i ^ j;
      if (ixj > i) {
        unsigned long long a = sCand[i], c = sCand[ixj];
        bool up = ((i & k) == 0);                 // descending overall
        if (up ? (a < c) : (a > c)) { sCand[i] = c; sCand[ixj] = a; }
      }
      __syncthreads();
    }
  }

  // ---- stage 6: decode boxes for candidates (SSD decode, VAR=0.1/0.2) ----
  {
    unsigned long long kv = sCand[tid];
    uint32_t key = (uint32_t)(kv >> 32);
    int valid = (tid < TOPK) && (key != 0u);
    sKeep[tid] = valid;
    if (valid) {
      int d = (int)(kv & 0xFFFFFFFFu);
      float4 l  = *(const float4*)(loc  + ((size_t)b * D_ + (size_t)d) * 4);
      float4 db = *(const float4*)(dbox + (size_t)d * 4);
      float cx = db.x + l.x * 0.1f * db.z;
      float cy = db.y + l.y * 0.1f * db.w;
      float w  = db.z * __expf(l.z * 0.2f);
      float h  = db.w * __expf(l.w * 0.2f);
      float x0 = cx - 0.5f * w, y0 = cy - 0.5f * h;
      sBox[tid][0] = x0; sBox[tid][1] = y0;
      sBox[tid][2] = x0 + w; sBox[tid][3] = y0 + h;
    }
  }
  __syncthreads();

  // ---- stage 7: greedy NMS (score order), parallel over j, serial over i ----
  for (int i = 0; i < TOPK; ++i) {
    if (sKeep[i]) {
      int j = tid;
      if (j > i && j < TOPK && sKeep[j]) {
        float ax0 = sBox[i][0], ay0 = sBox[i][1], ax1 = sBox[i][2], ay1 = sBox[i][3];
        float bx0 = sBox[j][0], by0 = sBox[j][1], bx1 = sBox[j][2], by1 = sBox[j][3];
        float tlx = fmaxf(ax0, bx0), tly = fmaxf(ay0, by0);
        float brx = fminf(ax1, bx1), bry = fminf(ay1, by1);
        float iw = fmaxf(brx - tlx, 0.0f), ih = fmaxf(bry - tly, 0.0f);
        float inter = iw * ih;
        float aA = (ax1 - ax0) * (ay1 - ay0);
        float aB = (bx1 - bx0) * (by1 - by0);
        float iou = inter / (aA + aB - inter + 1e-12f);
        if (iou > NMS_T) sKeep[j] = 0;
      }
    }
    __syncthreads();
  }

  // ---- stage 8: inclusive scan over keep, pack survivors ----
  sHist[tid] = (tid < TOPK) ? (uint32_t)sKeep[tid] : 0u;
  __syncthreads();
  for (int off = 1; off < NTHR; off <<= 1) {
    uint32_t v = sHist[tid];
    uint32_t a = (tid >= off) ? sHist[tid - off] : 0u;
    __syncthreads();
    sHist[tid] = v + a;
    __syncthreads();
  }
  if (tid < TOPK && sKeep[tid]) {
    int pos = (int)sHist[tid] - 1;
    unsigned long long kv = sCand[tid];
    float score = __uint_as_float((uint32_t)(kv >> 32));
    float* o = out + (((size_t)b * C_ + (size_t)cls) * TOPK + (size_t)pos) * 5;
    o[0] = score;
    o[1] = sBox[tid][0]; o[2] = sBox[tid][1];
    o[3] = sBox[tid][2]; o[4] = sBox[tid][3];
  }
}

// ---------------- host-side launch ----------------
extern "C" void kernel_launch(void* const* d_in, const int* in_sizes, int n_in,
                              void* d_out, int out_size, void* d_ws, size_t ws_size,
                              hipStream_t stream) {
  const float* loc  = (const float*)d_in[0];   // [B,D,4]
  const float* conf = (const float*)d_in[1];   // [B,D,C]
  const float* dbox = (const float*)d_in[2];   // [D,4]
  float* out = (float*)d_out;                  // [B,C,TOPK,5]

  float* mx     = (float*)d_ws;                // [B*D]
  float* invden = mx + (size_t)B_ * D_;        // [B*D]

  // background class + padding slots must be zero every call
  (void)hipMemsetAsync(d_out, 0, (size_t)out_size * sizeof(float), stream);

  int n = B_ * D_;
  ssd_stats_kernel<<<(n + NTHR - 1) / NTHR, NTHR, 0, stream>>>(conf, mx, invden, n);
  ssd_nms_kernel<<<B_ * (C_ - 1), NTHR, 0, stream>>>(conf, loc, dbox, mx, invden, out);
}